// HeteroRGCNLayer_18691697672931
// MI455X (gfx1250) — compile-verified
//
#include <hip/hip_runtime.h>
#include <hip/hip_bf16.h>

typedef __attribute__((ext_vector_type(2))) float v2f;
typedef __attribute__((ext_vector_type(8))) float v8f;

#define N_WORD 50000
#define N_DOC  50000
#define DIM    128
#define NE     600000
#define TILES_M (N_WORD / 16)        // 3125
#define TILES_N (DIM / 16)           // 8
#define TILES_PER_MAT (TILES_M * TILES_N)  // 25000
#define TOTAL_WAVES (3 * TILES_PER_MAT)    // 75000

// ---------------- zero init ----------------
__global__ void zero_f32_kernel(float* __restrict__ p, int n) {
    int i = blockIdx.x * blockDim.x + threadIdx.x;
    if (i < n) p[i] = 0.0f;
}

// ---------------- degree count (grid.y selects etype) ----------------
__global__ void degree_kernel(const int* __restrict__ dst_ww,
                              const int* __restrict__ dst_dw,
                              const int* __restrict__ dst_wd,
                              int* __restrict__ deg /* [3*N] contiguous */) {
    int e = blockIdx.x * blockDim.x + threadIdx.x;
    if (e >= NE) return;
    int et = blockIdx.y;
    const int* d = (et == 0) ? dst_ww : (et == 1 ? dst_dw : dst_wd);
    atomicAdd(&deg[et * N_WORD + d[e]], 1);   // emits global_atomic_add_u32
}

// ---------------- deg -> 1/max(deg,1), in place (int view -> float view) ----
__global__ void recip_kernel(int* __restrict__ deg, float* __restrict__ inv, int n) {
    int i = blockIdx.x * blockDim.x + threadIdx.x;
    if (i < n) {
        int d = deg[i];
        inv[i] = 1.0f / (float)(d > 1 ? d : 1);
    }
}

// ---------------- fused 3-matrix GEMM via V_WMMA_F32_16X16X4_F32 ----------
// out = [Wh_ww ; Wh_wd ; Wh_dw], each [50000,128], row-major.
// One wave computes one 16x16 tile; K=128 as 32 unrolled k=4 WMMA steps.
__global__ __launch_bounds__(256) void gemm_wmma_kernel(
    const float* __restrict__ h_word, const float* __restrict__ h_doc,
    const float* __restrict__ W0, const float* __restrict__ b0,
    const float* __restrict__ W1, const float* __restrict__ b1,
    const float* __restrict__ W2, const float* __restrict__ b2,
    float* __restrict__ out) {
    const int wid  = blockIdx.x * 8 + (threadIdx.x >> 5);   // 8 waves / 256-thread block
    const int lane = threadIdx.x & 31;
    if (wid >= TOTAL_WAVES) return;

    const int mat = wid / TILES_PER_MAT;
    const int rem = wid - mat * TILES_PER_MAT;
    const int mt  = rem >> 3;   // M tile 0..3124
    const int nt  = rem & 7;    // N tile 0..7

    const float* H = (mat == 2) ? h_doc : h_word;
    const float* W = (mat == 0) ? W0 : (mat == 1 ? W1 : W2);
    const float* B = (mat == 0) ? b0 : (mat == 1 ? b1 : b2);
    float* O = out + (size_t)mat * N_WORD * DIM;

    const int lr = lane & 15;   // M row (A) / N col (B,C,D) within tile
    const int lh = lane >> 4;   // K half selector for A/B fragments

    // C/D layout: vgpr i -> M = i + 8*lh, N = lr. Bias depends only on N.
    const float bias = B[nt * 16 + lr];
    v8f c;
#pragma unroll
    for (int i = 0; i < 8; ++i) c[i] = bias;

    // A 16x4 f32 layout: lanes0-15 {K0,K1}, lanes16-31 {K2,K3} -> float2 load.
    // B[k][n] = W[n][k] (torch W is [out,in]); same float2 pattern along k.
    const float* aptr = H + (size_t)(mt * 16 + lr) * DIM + lh * 2;
    const float* bptr = W + (size_t)(nt * 16 + lr) * DIM + lh * 2;
#pragma unroll
    for (int k0 = 0; k0 < DIM; k0 += 4) {
        v2f a = *(const v2f*)(aptr + k0);
        v2f b = *(const v2f*)(bptr + k0);
        c = __builtin_amdgcn_wmma_f32_16x16x4_f32(
            /*neg_a=*/false, a, /*neg_b=*/false, b,
            /*c_mod=*/(short)0, c, /*reuse_a=*/false, /*reuse_b=*/false);
    }

    float* optr = O + (size_t)(mt * 16 + lh * 8) * DIM + nt * 16 + lr;
#pragma unroll
    for (int i = 0; i < 8; ++i) optr[(size_t)i * DIM] = c[i];
}

// ---------------- edge scatter: acc[dst] += Wh[src] * inv_deg[dst] ---------
__global__ __launch_bounds__(256) void scatter_kernel(
    const float* __restrict__ Wh, const int* __restrict__ src,
    const int* __restrict__ dst, const float* __restrict__ inv,
    float* __restrict__ acc) {
    int t = blockIdx.x * 256 + threadIdx.x;
    int e = t >> 7;       // edge (2 edges per block)
    int f = t & 127;      // feature
    if (e >= NE) return;
    int s = src[e];
    int d = dst[e];
    float v = Wh[(size_t)s * DIM + f] * inv[d];
    unsafeAtomicAdd(&acc[(size_t)d * DIM + f], v);  // global_atomic_add_f32
}

// ---------------- in-place ReLU ----------------
__global__ void relu_kernel(float* __restrict__ p, int n) {
    int i = blockIdx.x * blockDim.x + threadIdx.x;
    if (i < n) {
        float v = p[i];
        p[i] = v > 0.0f ? v : 0.0f;
    }
}

extern "C" void kernel_launch(void* const* d_in, const int* in_sizes, int n_in,
                              void* d_out, int out_size, void* d_ws, size_t ws_size,
                              hipStream_t stream) {
    const float* h_word = (const float*)d_in[0];
    const float* h_doc  = (const float*)d_in[1];
    const float* W_ww   = (const float*)d_in[2];
    const float* b_ww   = (const float*)d_in[3];
    const float* W_wd   = (const float*)d_in[4];
    const float* b_wd   = (const float*)d_in[5];
    const float* W_dw   = (const float*)d_in[6];
    const float* b_dw   = (const float*)d_in[7];
    const int* src_ww = (const int*)d_in[8];
    const int* dst_ww = (const int*)d_in[9];
    const int* src_wd = (const int*)d_in[10];
    const int* dst_wd = (const int*)d_in[11];
    const int* src_dw = (const int*)d_in[12];
    const int* dst_dw = (const int*)d_in[13];

    float* out = (float*)d_out;   // [word 50000x128 ; doc 50000x128]

    // workspace layout
    float* Wh = (float*)d_ws;                                    // 3 x 50000 x 128 f32
    float* Wh_ww = Wh;
    float* Wh_wd = Wh + (size_t)1 * N_WORD * DIM;
    float* Wh_dw = Wh + (size_t)2 * N_WORD * DIM;
    int*   deg   = (int*)(Wh + (size_t)3 * N_WORD * DIM);        // 3 x 50000 ints
    float* inv   = (float*)deg;                                  // reused in place
    float* inv_ww = inv;
    float* inv_dw = inv + N_WORD;
    float* inv_wd = inv + 2 * N_WORD;

    // 1) zero accumulators (d_out) and degree counters
    {
        int n = out_size;
        zero_f32_kernel<<<(n + 255) / 256, 256, 0, stream>>>(out, n);
        int m = 3 * N_WORD;
        zero_f32_kernel<<<(m + 255) / 256, 256, 0, stream>>>((float*)deg, m);
    }

    // 2) degree counts (etype order in deg[]: ww, dw, wd)
    {
        dim3 grid((NE + 255) / 256, 3);
        degree_kernel<<<grid, 256, 0, stream>>>(dst_ww, dst_dw, dst_wd, deg);
    }

    // 3) reciprocal degrees
    {
        int m = 3 * N_WORD;
        recip_kernel<<<(m + 255) / 256, 256, 0, stream>>>(deg, inv, m);
    }

    // 4) WMMA GEMMs: Wh_ww, Wh_wd (from h_word), Wh_dw (from h_doc)
    {
        int blocks = (TOTAL_WAVES + 7) / 8;   // 8 waves per 256-thread block
        gemm_wmma_kernel<<<blocks, 256, 0, stream>>>(
            h_word, h_doc, W_ww, b_ww, W_wd, b_wd, W_dw, b_dw, Wh);
    }

    // 5) scatter-add per etype (word dsts get ww + dw; doc dsts get wd)
    {
        int blocks = (NE * 128) / 256;   // 300000
        scatter_kernel<<<blocks, 256, 0, stream>>>(Wh_ww, src_ww, dst_ww, inv_ww, out);
        scatter_kernel<<<blocks, 256, 0, stream>>>(Wh_dw, src_dw, dst_dw, inv_dw, out);
        scatter_kernel<<<blocks, 256, 0, stream>>>(Wh_wd, src_wd, dst_wd, inv_wd,
                                                   out + (size_t)N_WORD * DIM);
    }

    // 6) ReLU in place
    relu_kernel<<<(out_size + 255) / 256, 256, 0, stream>>>(out, out_size);
}